// FrozenSTGAEEncoder_47132971107177
// MI455X (gfx1250) — compile-verified
//
#include <hip/hip_runtime.h>
#include <hip/hip_bf16.h>
#include <math.h>

// ---------------------------------------------------------------------------
// FrozenSTGAEEncoder for MI455X (gfx1250, wave32, fp32 WMMA)
//
//  gcn_features = tanh( x_flat @ Wbig + bbig ),  Wbig[(m,f),(n,h)] folds the
//  5x5 normalized adjacency into the GCN weight:  [B*T,52] x [52,320]
//  GRU: 72 sequential steps of two GEMMs + fused gate kernel.
//
//  GEMM: each wave owns a 16x64 C strip (4 WMMA tiles), K in chunks of 4 via
//  V_WMMA_F32_16X16X4_F32. B matrices are pre-packed into fragment order so
//  the inner loop is:  1 float2 A load + 4 float2 B loads + 4 WMMAs, no
//  branches, no EXEC games.
// ---------------------------------------------------------------------------

typedef float v2f __attribute__((ext_vector_type(2)));
typedef float v8f __attribute__((ext_vector_type(8)));

#define N_NODES 5
#define IN_FEAT 10
#define GCN_HID 64
#define GRU_HID 256
#define BATCH   2048
#define SEQ     72
#define N_EDGES 16

#define GCN_COLS (N_NODES * GCN_HID)     // 320
#define GCN_K    (N_NODES * IN_FEAT)     // 50
#define GCN_KPAD 52                      // padded to multiple of 4
#define ROWS_BT  (BATCH * SEQ)           // 147456
#define G3H      (3 * GRU_HID)           // 768

static __device__ __forceinline__ v8f wmma4(v2f a, v2f b, v8f c) {
    return __builtin_amdgcn_wmma_f32_16x16x4_f32(
        false, a, false, b, (short)0, c, false, false);
}

// ---------------------------------------------------------------------------
// Prep: A_norm (5x5) from edge_index -> Wbig [52][320] K-major (rows 50..51
// zero) and bbig[320].
// ---------------------------------------------------------------------------
__global__ void prep_gcn_kernel(const int* __restrict__ ei,
                                const float* __restrict__ gcn_w,  // [10][64]
                                const float* __restrict__ gcn_b,  // [64]
                                float* __restrict__ Wbig,         // [52][320]
                                float* __restrict__ bbig)         // [320]
{
    __shared__ float A[N_NODES][N_NODES];
    if (threadIdx.x == 0) {
        float deg[N_NODES];
        float Aa[N_NODES][N_NODES];
        for (int n = 0; n < N_NODES; ++n) {
            deg[n] = 1.0f;   // self loop
            for (int m = 0; m < N_NODES; ++m) Aa[n][m] = 0.0f;
        }
        for (int e = 0; e < N_EDGES; ++e) deg[ei[N_EDGES + e]] += 1.0f;
        float dinv[N_NODES];
        for (int n = 0; n < N_NODES; ++n) dinv[n] = rsqrtf(deg[n]);
        for (int e = 0; e < N_EDGES; ++e) {
            int s = ei[e], d = ei[N_EDGES + e];
            Aa[d][s] += dinv[s] * dinv[d];
        }
        for (int n = 0; n < N_NODES; ++n) Aa[n][n] += dinv[n] * dinv[n];
        for (int n = 0; n < N_NODES; ++n)
            for (int m = 0; m < N_NODES; ++m) A[n][m] = Aa[n][m];
    }
    __syncthreads();
    int col = threadIdx.x;                 // 0..319
    if (col < GCN_COLS) {
        int n = col / GCN_HID;
        int hh = col % GCN_HID;
        for (int k = 0; k < GCN_KPAD; ++k) {
            float v = 0.0f;
            if (k < GCN_K) {
                int m = k / IN_FEAT, f = k % IN_FEAT;
                v = A[n][m] * gcn_w[f * GCN_HID + hh];
            }
            Wbig[k * GCN_COLS + col] = v;
        }
        bbig[col] = gcn_b[hh];
    }
}

// dst[cols][rows] = src[rows][cols]^T
__global__ void transpose_kernel(float* __restrict__ dst,
                                 const float* __restrict__ src,
                                 int rows, int cols)
{
    int i = blockIdx.x * blockDim.x + threadIdx.x;
    if (i < rows * cols) {
        int r = i / cols, c = i % cols;
        dst[c * rows + r] = src[i];
    }
}

__global__ void zero_kernel(float* __restrict__ p, int n)
{
    int i = blockIdx.x * blockDim.x + threadIdx.x;
    if (i < n) p[i] = 0.0f;
}

// xpad [rows][52] <- x [rows][50], zero-padded K tail
__global__ void pad_x_kernel(float* __restrict__ dst,
                             const float* __restrict__ src, int rows)
{
    int i = blockIdx.x * blockDim.x + threadIdx.x;
    if (i < rows * GCN_KPAD) {
        int r = i / GCN_KPAD, c = i % GCN_KPAD;
        dst[i] = (c < GCN_K) ? src[r * GCN_K + c] : 0.0f;
    }
}

// ---------------------------------------------------------------------------
// Pack a K-major B matrix [Kpad][N] into WMMA fragment order:
//   dst2[ ((g*numKc + kc)*4 + j)*32 + lane ] =
//       { B[kc*4 + kOff][col], B[kc*4 + kOff + 1][col] }
//   col = g*64 + j*16 + (lane&15),  kOff = (lane>>4)*2
// One wave's per-chunk fragment is then a single contiguous float2 load.
// ---------------------------------------------------------------------------
__global__ void pack_bfrag_kernel(v2f* __restrict__ dst,
                                  const float* __restrict__ src,
                                  int N, int numKc)
{
    int total = (N >> 6) * numKc * 4 * 32;   // number of float2 elements
    int i = blockIdx.x * blockDim.x + threadIdx.x;
    if (i >= total) return;
    int lane = i & 31;
    int j    = (i >> 5) & 3;
    int kc   = (i >> 7) % numKc;
    int g    = i / (numKc << 7);
    int col  = (g << 6) + (j << 4) + (lane & 15);
    int k    = (kc << 2) + ((lane >> 4) << 1);
    v2f v;
    v.x = src[(size_t)k * N + col];
    v.y = src[(size_t)(k + 1) * N + col];
    dst[i] = v;
}

// ---------------------------------------------------------------------------
// fp32 WMMA GEMM, 16x64 strip per wave, K multiple of 4, branch-free.
//   C[m][n] = act( sum_k A[m][k]*B[k][n] + bias[n] )
// ---------------------------------------------------------------------------
__global__ __launch_bounds__(256)
void wmma_gemm_f32(const float* __restrict__ Amat, int lda,
                   const v2f* __restrict__ Bfrag, int N, int numKc,
                   const float* __restrict__ bias,
                   float* __restrict__ C, int ldc,
                   int M, int act_tanh)
{
    int wave = threadIdx.x >> 5;
    int lane = threadIdx.x & 31;
    int colGroups = N >> 6;                       // 64 cols per wave strip
    int numTiles  = (M >> 4) * colGroups;
    int tile = blockIdx.x * 8 + wave;
    if (tile >= numTiles) return;                 // wave-uniform branch

    int rowBase = (tile / colGroups) << 4;
    int g       = tile % colGroups;
    int l16  = lane & 15;
    int kOff = (lane >> 4) << 1;                  // 0 | 2

    const float* Arow = Amat + (size_t)(rowBase + l16) * lda + kOff;
    const v2f*   bptr = Bfrag + ((size_t)g * numKc << 7) + lane;

    v8f acc0 = {}, acc1 = {}, acc2 = {}, acc3 = {};
    for (int kc = 0; kc < numKc; ++kc) {
        v2f a  = *(const v2f*)(Arow + (kc << 2));
        v2f b0 = bptr[0];
        v2f b1 = bptr[32];
        v2f b2 = bptr[64];
        v2f b3 = bptr[96];
        bptr += 128;
        acc0 = wmma4(a, b0, acc0);
        acc1 = wmma4(a, b1, acc1);
        acc2 = wmma4(a, b2, acc2);
        acc3 = wmma4(a, b3, acc3);
    }

    int colBase = g << 6;
    float bv0 = bias[colBase + l16];
    float bv1 = bias[colBase + 16 + l16];
    float bv2 = bias[colBase + 32 + l16];
    float bv3 = bias[colBase + 48 + l16];
    int rAdd = (lane >> 4) << 3;                  // +8 rows for lanes 16-31
    #pragma unroll
    for (int r = 0; r < 8; ++r) {
        size_t o = (size_t)(rowBase + r + rAdd) * ldc + colBase + l16;
        float v0 = acc0[r] + bv0;
        float v1 = acc1[r] + bv1;
        float v2 = acc2[r] + bv2;
        float v3 = acc3[r] + bv3;
        if (act_tanh) {
            v0 = tanhf(v0); v1 = tanhf(v1); v2 = tanhf(v2); v3 = tanhf(v3);
        }
        C[o]      = v0;
        C[o + 16] = v1;
        C[o + 32] = v2;
        C[o + 48] = v3;
    }
}

// ---------------------------------------------------------------------------
// GRU gate fusion: h_new = (1-z)*n + z*h_prev   (in-place over h is safe)
// ---------------------------------------------------------------------------
__global__ void gru_gate_kernel(const float* __restrict__ xproj, // [B][768]
                                const float* __restrict__ gh,    // [B][768]
                                const float* __restrict__ hprev, // [B][256]
                                float* __restrict__ hout,        // [B][256]
                                int total)
{
    int i = blockIdx.x * blockDim.x + threadIdx.x;
    if (i >= total) return;
    int b = i / GRU_HID, c = i % GRU_HID;
    const float* xp = xproj + (size_t)b * G3H;
    const float* g  = gh    + (size_t)b * G3H;
    float r = 1.0f / (1.0f + expf(-(xp[c] + g[c])));
    float z = 1.0f / (1.0f + expf(-(xp[c + GRU_HID] + g[c + GRU_HID])));
    float n = tanhf(xp[c + 2 * GRU_HID] + r * g[c + 2 * GRU_HID]);
    hout[i] = (1.0f - z) * n + z * hprev[i];
}

// ---------------------------------------------------------------------------
extern "C" void kernel_launch(void* const* d_in, const int* in_sizes, int n_in,
                              void* d_out, int out_size, void* d_ws, size_t ws_size,
                              hipStream_t stream)
{
    (void)in_sizes; (void)n_in; (void)out_size; (void)ws_size;

    const float* x      = (const float*)d_in[0];   // [2048,72,5,10]
    const int*   eidx   = (const int*)  d_in[1];   // [2,16]
    const float* gcn_w  = (const float*)d_in[2];   // [10,64]
    const float* gcn_b  = (const float*)d_in[3];   // [64]
    const float* w_ih   = (const float*)d_in[4];   // [768,320]
    const float* w_hh   = (const float*)d_in[5];   // [768,256]
    const float* b_ih   = (const float*)d_in[6];   // [768]
    const float* b_hh   = (const float*)d_in[7];   // [768]

    float* out_gcn = (float*)d_out;                              // [147456,320]
    float* out_h   = out_gcn + (size_t)ROWS_BT * GCN_COLS;       // [2048,256]

    // workspace layout (floats)
    float* ws      = (float*)d_ws;
    float* Wbig    = ws;                            // 52*320   = 16640
    float* bbig    = Wbig   + GCN_KPAD * GCN_COLS;  // 320
    float* wihT    = bbig   + GCN_COLS;             // 320*768  = 245760
    float* whhT    = wihT   + GCN_COLS * G3H;       // 256*768  = 196608
    float* fragGCN = whhT   + GRU_HID * G3H;        // 16640
    float* fragIH  = fragGCN + GCN_KPAD * GCN_COLS; // 245760
    float* fragHH  = fragIH + GCN_COLS * G3H;       // 196608
    float* hbuf    = fragHH + GRU_HID * G3H;        // 2048*256 = 524288
    float* xproj   = hbuf   + BATCH * GRU_HID;      // 2048*768 = 1572864
    float* ghbuf   = xproj  + BATCH * G3H;          // 2048*768 = 1572864
    float* xpad    = ghbuf  + BATCH * G3H;          // 147456*52 = 7667712

    const int nKcGCN = GCN_KPAD / 4;   // 13
    const int nKcIH  = GCN_COLS / 4;   // 80
    const int nKcHH  = GRU_HID  / 4;   // 64

    // --- prep -------------------------------------------------------------
    hipLaunchKernelGGL(prep_gcn_kernel, dim3(1), dim3(320), 0, stream,
                       eidx, gcn_w, gcn_b, Wbig, bbig);
    hipLaunchKernelGGL(transpose_kernel, dim3((G3H * GCN_COLS + 255) / 256),
                       dim3(256), 0, stream, wihT, w_ih, G3H, GCN_COLS);
    hipLaunchKernelGGL(transpose_kernel, dim3((G3H * GRU_HID + 255) / 256),
                       dim3(256), 0, stream, whhT, w_hh, G3H, GRU_HID);

    hipLaunchKernelGGL(pack_bfrag_kernel,
                       dim3((GCN_KPAD * GCN_COLS / 2 + 255) / 256), dim3(256),
                       0, stream, (v2f*)fragGCN, Wbig, GCN_COLS, nKcGCN);
    hipLaunchKernelGGL(pack_bfrag_kernel,
                       dim3((GCN_COLS * G3H / 2 + 255) / 256), dim3(256),
                       0, stream, (v2f*)fragIH, wihT, G3H, nKcIH);
    hipLaunchKernelGGL(pack_bfrag_kernel,
                       dim3((GRU_HID * G3H / 2 + 255) / 256), dim3(256),
                       0, stream, (v2f*)fragHH, whhT, G3H, nKcHH);

    hipLaunchKernelGGL(zero_kernel, dim3((BATCH * GRU_HID + 255) / 256),
                       dim3(256), 0, stream, hbuf, BATCH * GRU_HID);
    hipLaunchKernelGGL(pad_x_kernel, dim3((ROWS_BT * GCN_KPAD + 255) / 256),
                       dim3(256), 0, stream, xpad, x, ROWS_BT);

    // --- GCN GEMM: [147456,52] x [52,320] -> tanh -> out_gcn ---------------
    {
        int numTiles = (ROWS_BT / 16) * (GCN_COLS / 64);   // 9216*5
        int blocks   = (numTiles + 7) / 8;
        hipLaunchKernelGGL(wmma_gemm_f32, dim3(blocks), dim3(256), 0, stream,
                           xpad, GCN_KPAD, (const v2f*)fragGCN, GCN_COLS,
                           nKcGCN, bbig, out_gcn, GCN_COLS, ROWS_BT, 1);
    }

    // --- GRU: 72 sequential steps -----------------------------------------
    int gruTiles  = (BATCH / 16) * (G3H / 64);             // 128*12 = 1536
    int gruBlocks = (gruTiles + 7) / 8;                    // 192
    int gateTotal = BATCH * GRU_HID;
    for (int t = 0; t < SEQ; ++t) {
        // xproj = gru_in[:,t,:] @ wihT + b_ih   (A row stride 72*320)
        hipLaunchKernelGGL(wmma_gemm_f32, dim3(gruBlocks), dim3(256), 0, stream,
                           out_gcn + (size_t)t * GCN_COLS, SEQ * GCN_COLS,
                           (const v2f*)fragIH, G3H, nKcIH, b_ih,
                           xproj, G3H, BATCH, 0);
        // gh = h @ whhT + b_hh
        hipLaunchKernelGGL(wmma_gemm_f32, dim3(gruBlocks), dim3(256), 0, stream,
                           hbuf, GRU_HID,
                           (const v2f*)fragHH, G3H, nKcHH, b_hh,
                           ghbuf, G3H, BATCH, 0);
        // gates + state update (last step writes straight to d_out tail)
        float* hdst = (t == SEQ - 1) ? out_h : hbuf;
        hipLaunchKernelGGL(gru_gate_kernel, dim3((gateTotal + 255) / 256),
                           dim3(256), 0, stream,
                           xproj, ghbuf, hbuf, hdst, gateTotal);
    }
}